// QMLP_21165598835258
// MI455X (gfx1250) — compile-verified
//
#include <hip/hip_runtime.h>
#include <math.h>

typedef float v2f __attribute__((ext_vector_type(2)));
typedef float v4f __attribute__((ext_vector_type(4)));
typedef float v8f __attribute__((ext_vector_type(8)));

struct c32 { float x, y; };

// ---------------------------------------------------------------------------
// Kernel 1: 4x32x32 mean-pool of x (128,2,16,128,128) -> p (128,128)
// Memory-bound: 256 MB streamed once -> wide b128 non-temporal loads.
// Grid: 1024 blocks = (b, c, t_outer); 256 threads = 8 waves.
// Each wave reads full 512B rows (32 lanes x float4), ho loop is static so
// accumulators stay in registers.
// ---------------------------------------------------------------------------
__global__ __launch_bounds__(256) void pool_kernel(const float* __restrict__ x,
                                                   float* __restrict__ p) {
    const int blk = blockIdx.x;
    const int to = blk & 3;
    const int c  = (blk >> 2) & 1;
    const int b  = blk >> 3;
    const float* base = x + (size_t)(((b * 2 + c) * 16 + to * 4)) * (128 * 128);

    const int tid  = threadIdx.x;
    const int lane = tid & 31;
    const int wave = tid >> 5;

    __shared__ float lds[8][32][4];

    #pragma unroll
    for (int ho = 0; ho < 4; ++ho) {
        float a = 0.0f;
        #pragma unroll
        for (int ti = 0; ti < 4; ++ti) {
            for (int hh = wave; hh < 32; hh += 8) {
                const int h = ho * 32 + hh;
                v4f v = __builtin_nontemporal_load(
                    (const v4f*)(base + ti * 16384 + h * 128 + lane * 4));
                a += v.x + v.y + v.z + v.w;
            }
        }
        lds[wave][lane][ho] = a;
    }
    __syncthreads();

    if (tid < 16) {
        const int ho = tid >> 2;
        const int wo = tid & 3;
        float s = 0.0f;
        #pragma unroll
        for (int w = 0; w < 8; ++w)
            #pragma unroll
            for (int l = 0; l < 8; ++l)
                s += lds[w][wo * 8 + l][ho];
        p[((b * 2 + c) * 4 + to) * 16 + ho * 4 + wo] = s * (1.0f / 4096.0f);
    }
}

// ---------------------------------------------------------------------------
// Kernel 2: MLP head via f32 WMMA.
// GEMM1: h = relu(F(128x128) @ w1^T(128x64) + b1) with V_WMMA_F32_16X16X4_F32.
//   1 block, 8 waves; wave w owns M-tile w (rows 16w..16w+15), all 4 N-tiles.
//   32 K-steps of K=4 -> 128 wmma per wave.
// GEMM2: qin = tanh(h(128x64) @ w2^T(64x8) + b2) * pi  (tiny, plain VALU).
// A fragment (16x4): lane<16 -> K={k0,k0+1}; lane>=16 -> K={k0+2,k0+3}.
// C/D: vgpr j -> M = j (+8 for upper lane half), N = lane&15.
// ---------------------------------------------------------------------------
__global__ __launch_bounds__(256) void mlp_kernel(const float* __restrict__ F,
                                                  const float* __restrict__ w1,
                                                  const float* __restrict__ b1,
                                                  const float* __restrict__ w2,
                                                  const float* __restrict__ b2,
                                                  float* __restrict__ qin) {
    __shared__ float h[128][64];

    const int tid  = threadIdx.x;
    const int lane = tid & 31;
    const int wave = tid >> 5;
    const int lr   = lane & 15;   // M (for A) / N (for B,C,D)
    const int hi   = lane >> 4;   // half-wave selector
    const int mbase = wave * 16;

    v8f acc[4] = {};

    for (int k0 = 0; k0 < 128; k0 += 4) {
        const int k = k0 + hi * 2;
        v2f a = *(const v2f*)(F + (mbase + lr) * 128 + k);
        #pragma unroll
        for (int nt = 0; nt < 4; ++nt) {
            // B[k][n] = w1[n][k]  -> row-major w1 gives an aligned float2
            v2f bf = *(const v2f*)(w1 + (nt * 16 + lr) * 128 + k);
            acc[nt] = __builtin_amdgcn_wmma_f32_16x16x4_f32(
                false, a, false, bf, (short)0, acc[nt], false, false);
        }
    }

    #pragma unroll
    for (int nt = 0; nt < 4; ++nt) {
        const int n = nt * 16 + lr;
        const float bias = b1[n];
        #pragma unroll
        for (int j = 0; j < 8; ++j) {
            const int m = mbase + j + hi * 8;
            const float v = acc[nt][j] + bias;
            h[m][n] = v > 0.0f ? v : 0.0f;
        }
    }
    __syncthreads();

    // qin[b][o] = tanh(h[b][:] . w2[o][:] + b2[o]) * pi   (128x8 outputs)
    for (int idx = tid; idx < 1024; idx += 256) {
        const int bb = idx >> 3;
        const int o  = idx & 7;
        float s = b2[o];
        const float* wrow = w2 + o * 64;
        #pragma unroll 8
        for (int k = 0; k < 64; ++k) s += h[bb][k] * wrow[k];
        qin[idx] = tanhf(s) * 3.14159265358979323846f;
    }
}

// ---------------------------------------------------------------------------
// Kernel 3: quantum circuit + output heads.
// One block per batch element; 256 complex amplitudes live in LDS, one per
// thread. Gate updates are disjoint-pair read/modify/writes (no races), one
// barrier per gate. Fully deterministic fixed-order reductions.
// Qubit i maps to bit (7-i) of the basis index (jnp C-order flatten).
// ---------------------------------------------------------------------------
__device__ __forceinline__ void apply1q(c32* st, int tid, int wire,
                                        c32 u00, c32 u01, c32 u10, c32 u11) {
    if (tid < 128) {
        const unsigned mask = 1u << (7 - wire);
        const unsigned low  = (unsigned)tid & (mask - 1u);
        const unsigned s0 = (((unsigned)tid & ~(mask - 1u)) << 1) | low;
        const unsigned s1 = s0 | mask;
        const c32 a0 = st[s0], a1 = st[s1];
        c32 n0, n1;
        n0.x = u00.x * a0.x - u00.y * a0.y + u01.x * a1.x - u01.y * a1.y;
        n0.y = u00.x * a0.y + u00.y * a0.x + u01.x * a1.y + u01.y * a1.x;
        n1.x = u10.x * a0.x - u10.y * a0.y + u11.x * a1.x - u11.y * a1.y;
        n1.y = u10.x * a0.y + u10.y * a0.x + u11.x * a1.y + u11.y * a1.x;
        st[s0] = n0; st[s1] = n1;
    }
    __syncthreads();
}

__device__ __forceinline__ void apply_crx(c32* st, int tid, int ctrl, int tgt,
                                          float c, float s) {
    if (tid < 64) {
        const unsigned mC = 1u << (7 - ctrl);
        const unsigned mT = 1u << (7 - tgt);
        const unsigned lo = mC < mT ? mC : mT;
        const unsigned hg = mC < mT ? mT : mC;
        unsigned v = (unsigned)tid;
        v = ((v & ~(lo - 1u)) << 1) | (v & (lo - 1u));
        v = ((v & ~(hg - 1u)) << 1) | (v & (hg - 1u));
        v |= mC;                       // control bit set; target bit is 0
        const unsigned s0 = v;
        const unsigned s1 = v | mT;
        const c32 a0 = st[s0], a1 = st[s1];
        // RX: n0 = c*a0 - i*s*a1 ; n1 = -i*s*a0 + c*a1
        c32 n0, n1;
        n0.x = c * a0.x + s * a1.y;  n0.y = c * a0.y - s * a1.x;
        n1.x = c * a1.x + s * a0.y;  n1.y = c * a1.y - s * a0.x;
        st[s0] = n0; st[s1] = n1;
    }
    __syncthreads();
}

__global__ __launch_bounds__(256) void quantum_kernel(
    const float* __restrict__ qin, const float* __restrict__ qw,
    const float* __restrict__ wave_w, const float* __restrict__ wave_b,
    const float* __restrict__ volt_w, const float* __restrict__ volt_b,
    float* __restrict__ out) {
    const int b   = blockIdx.x;
    const int tid = threadIdx.x;

    __shared__ c32  st[256];
    __shared__ float rxc[8], rxs[8];
    __shared__ c32  rot[2][8][4];      // m00,m01,m10,m11
    __shared__ float crc[2][8], crs[2][8];
    __shared__ float probs[256];
    __shared__ float qo[8];

    st[tid].x = (tid == 0) ? 1.0f : 0.0f;
    st[tid].y = 0.0f;

    if (tid < 8) {
        const float th = qin[b * 8 + tid];
        rxc[tid] = cosf(0.5f * th);
        rxs[tid] = sinf(0.5f * th);
    }
    if (tid < 16) {
        const int layer = tid >> 3, i = tid & 7;
        const float phi = qw[(layer * 8 + i) * 3 + 0];
        const float th  = qw[(layer * 8 + i) * 3 + 1];
        const float om  = qw[(layer * 8 + i) * 3 + 2];
        const float ct = cosf(0.5f * th), sn = sinf(0.5f * th);
        const float ap = 0.5f * (phi + om), am = 0.5f * (phi - om);
        rot[layer][i][0] = {  cosf(ap) * ct, -sinf(ap) * ct };  // e^{-i ap} c
        rot[layer][i][1] = { -cosf(am) * sn, -sinf(am) * sn };  // -e^{+i am} s
        rot[layer][i][2] = {  cosf(am) * sn, -sinf(am) * sn };  // e^{-i am} s
        rot[layer][i][3] = {  cosf(ap) * ct,  sinf(ap) * ct };  // e^{+i ap} c
        crc[layer][i] = cosf(0.5f * phi);
        crs[layer][i] = sinf(0.5f * phi);
    }
    __syncthreads();

    for (int layer = 0; layer < 2; ++layer) {
        for (int i = 0; i < 8; ++i) {
            c32 d  = { rxc[i], 0.0f };
            c32 od = { 0.0f, -rxs[i] };
            apply1q(st, tid, i, d, od, od, d);
        }
        for (int i = 0; i < 8; ++i)
            apply1q(st, tid, i, rot[layer][i][0], rot[layer][i][1],
                                rot[layer][i][2], rot[layer][i][3]);
        for (int i = 0; i < 8; ++i)
            apply_crx(st, tid, i, (i + 1) & 7, crc[layer][i], crs[layer][i]);
    }

    probs[tid] = st[tid].x * st[tid].x + st[tid].y * st[tid].y;
    __syncthreads();

    if (tid < 8) {
        const int shift = 7 - tid;
        float s = 0.0f;
        for (int k = 0; k < 256; ++k) {
            const float p = probs[k];
            s += ((k >> shift) & 1) ? -p : p;
        }
        qo[tid] = s;
    }
    __syncthreads();

    if (tid < 4) {
        float s = wave_b[tid];
        #pragma unroll
        for (int j = 0; j < 8; ++j) s += qo[j] * wave_w[tid * 8 + j];
        out[b * 4 + tid] = s;                  // wave_out
    } else if (tid < 8) {
        const int t = tid - 4;
        float s = volt_b[t];
        #pragma unroll
        for (int j = 0; j < 8; ++j) s += qo[j] * volt_w[t * 8 + j];
        out[512 + b * 4 + t] = s;              // volt_out
    }
}

// ---------------------------------------------------------------------------
extern "C" void kernel_launch(void* const* d_in, const int* in_sizes, int n_in,
                              void* d_out, int out_size, void* d_ws, size_t ws_size,
                              hipStream_t stream) {
    const float* x      = (const float*)d_in[0];
    const float* w1     = (const float*)d_in[1];
    const float* b1     = (const float*)d_in[2];
    const float* w2     = (const float*)d_in[3];
    const float* b2     = (const float*)d_in[4];
    const float* qw     = (const float*)d_in[5];
    const float* wave_w = (const float*)d_in[6];
    const float* wave_b = (const float*)d_in[7];
    const float* volt_w = (const float*)d_in[8];
    const float* volt_b = (const float*)d_in[9];

    float* p   = (float*)d_ws;     // 128*128 floats
    float* qin = p + 128 * 128;    // 128*8 floats

    pool_kernel<<<1024, 256, 0, stream>>>(x, p);
    mlp_kernel<<<1, 256, 0, stream>>>(p, w1, b1, w2, b2, qin);
    quantum_kernel<<<128, 256, 0, stream>>>(qin, qw, wave_w, wave_b, volt_w,
                                            volt_b, (float*)d_out);
}